// MinibatchDiscrimination1d_2035814498340
// MI455X (gfx1250) — compile-verified
//
#include <hip/hip_runtime.h>

// ---------------------------------------------------------------------------
// Problem constants (from reference): x (512,1024) f32, T (1024,64,16) f32
//   M   = x @ T.reshape(1024,1024)            -> (512,1024) in d_ws
//   out[i,b] = sum_j exp(-sum_c |M[i,b*16+c]-M[j,b*16+c]|) - 1
//   d_out = concat([x, out], axis=1)          -> (512, 1088) f32
// ---------------------------------------------------------------------------
#define NN    512     // batch
#define KA    1024    // A (reduction dim)
#define NB    64      // B
#define NC    16      // C
#define BC    1024    // B*C (GEMM N dim)
#define OUTW  1088    // A + B

typedef float v2f __attribute__((ext_vector_type(2)));
typedef float v8f __attribute__((ext_vector_type(8)));

// ---------------------------------------------------------------------------
// Kernel 1: f32 GEMM via V_WMMA_F32_16X16X4_F32.
// Each wave owns a 16x64 output tile: one A fragment, four B fragments,
// four v8f accumulators. Grid: 64 blocks x 8 waves = 512 wave-tiles
// = (512/16 M-tiles) x (1024/64 N-tiles).
// ---------------------------------------------------------------------------
__global__ __launch_bounds__(256) void gemm_f32_wmma(
    const float* __restrict__ x,   // (512,1024)
    const float* __restrict__ W,   // (1024,1024) = T reshaped
    float* __restrict__ M)         // (512,1024)
{
  const int lane = threadIdx.x & 31;
  const int wave = threadIdx.x >> 5;
  const int gw   = blockIdx.x * 8 + wave;      // 0..511
  const int tm   = gw >> 4;                    // M tile 0..31
  const int tn   = gw & 15;                    // N wave-tile 0..15
  const int row0 = tm * 16;
  const int col0 = tn * 64;
  const int half = lane >> 4;                  // 0: lanes 0-15, 1: lanes 16-31
  const int l15  = lane & 15;

  v8f c0 = {}, c1 = {}, c2 = {}, c3 = {};

  // A fragment (ISA 32-bit A 16x4): lane l15 = row, K = vgpr + 2*half
  const float* arow = x + (row0 + l15) * KA;

  for (int k = 0; k < KA; k += 4) {
    // A: two consecutive K values -> single 8B load (8B aligned: k%4==0)
    v2f a = *(const v2f*)(arow + k + 2 * half);

    // B fragments (4x16, K = vgpr + 2*half, N = l15)
    const float* bp = W + (k + 2 * half) * BC + col0 + l15;
    v2f b0, b1, b2, b3;
    b0.x = bp[0];  b0.y = bp[BC];
    b1.x = bp[16]; b1.y = bp[BC + 16];
    b2.x = bp[32]; b2.y = bp[BC + 32];
    b3.x = bp[48]; b3.y = bp[BC + 48];

    c0 = __builtin_amdgcn_wmma_f32_16x16x4_f32(false, a, false, b0, (short)0, c0, false, false);
    c1 = __builtin_amdgcn_wmma_f32_16x16x4_f32(false, a, false, b1, (short)0, c1, false, false);
    c2 = __builtin_amdgcn_wmma_f32_16x16x4_f32(false, a, false, b2, (short)0, c2, false, false);
    c3 = __builtin_amdgcn_wmma_f32_16x16x4_f32(false, a, false, b3, (short)0, c3, false, false);
  }

  // C/D layout (ISA 32-bit 16x16): N = l15, M = vgpr + 8*half
  #pragma unroll
  for (int r = 0; r < 8; ++r) {
    float* mrow = M + (row0 + r + 8 * half) * BC + col0 + l15;
    mrow[0]  = c0[r];
    mrow[16] = c1[r];
    mrow[32] = c2[r];
    mrow[48] = c3[r];
  }
}

// ---------------------------------------------------------------------------
// Kernel 2: pairwise exp(-L1) reduction. One block per b (64 blocks).
// Stage the 512x16 slice (32KB) in LDS; every lane of a wave reads the same
// S[j] row per iteration -> LDS broadcast, conflict-free.
// Writes results straight into d_out column block [1024..1087].
// ---------------------------------------------------------------------------
__global__ __launch_bounds__(256) void pairwise_exp_l1(
    const float* __restrict__ M,   // (512,1024)
    float* __restrict__ out)       // (512,1088)
{
  const int b = blockIdx.x;        // 0..63
  __shared__ float S[NN * NC];     // 32 KB

  for (int idx = threadIdx.x; idx < NN * NC; idx += blockDim.x) {
    const int i = idx >> 4, c = idx & 15;
    S[idx] = M[i * BC + b * NC + c];
  }
  __syncthreads();

  for (int i = threadIdx.x; i < NN; i += blockDim.x) {
    const float4 r0 = *(const float4*)&S[i * NC + 0];
    const float4 r1 = *(const float4*)&S[i * NC + 4];
    const float4 r2 = *(const float4*)&S[i * NC + 8];
    const float4 r3 = *(const float4*)&S[i * NC + 12];
    float acc = 0.0f;
    for (int j = 0; j < NN; ++j) {
      const float4 q0 = *(const float4*)&S[j * NC + 0];
      const float4 q1 = *(const float4*)&S[j * NC + 4];
      const float4 q2 = *(const float4*)&S[j * NC + 8];
      const float4 q3 = *(const float4*)&S[j * NC + 12];
      float l1 = fabsf(r0.x - q0.x) + fabsf(r0.y - q0.y) +
                 fabsf(r0.z - q0.z) + fabsf(r0.w - q0.w);
      l1 += fabsf(r1.x - q1.x) + fabsf(r1.y - q1.y) +
            fabsf(r1.z - q1.z) + fabsf(r1.w - q1.w);
      l1 += fabsf(r2.x - q2.x) + fabsf(r2.y - q2.y) +
            fabsf(r2.z - q2.z) + fabsf(r2.w - q2.w);
      l1 += fabsf(r3.x - q3.x) + fabsf(r3.y - q3.y) +
            fabsf(r3.z - q3.z) + fabsf(r3.w - q3.w);
      acc += __expf(-l1);          // v_exp_f32 path
    }
    out[i * OUTW + KA + b] = acc - 1.0f;  // subtract the j==i self term
  }
}

// ---------------------------------------------------------------------------
// Kernel 3: copy x into output columns [0..1023].
// ---------------------------------------------------------------------------
__global__ __launch_bounds__(256) void copy_x(
    const float* __restrict__ x, float* __restrict__ out)
{
  const int idx = blockIdx.x * blockDim.x + threadIdx.x;  // 0 .. 512*1024-1
  const int i = idx >> 10, a = idx & 1023;
  out[i * OUTW + a] = x[idx];
}

// ---------------------------------------------------------------------------
extern "C" void kernel_launch(void* const* d_in, const int* in_sizes, int n_in,
                              void* d_out, int out_size, void* d_ws, size_t ws_size,
                              hipStream_t stream) {
  const float* x = (const float*)d_in[0];   // 512*1024
  const float* T = (const float*)d_in[1];   // 1024*64*16 (row-major == (1024,1024))
  float* out = (float*)d_out;               // 512*1088
  float* M   = (float*)d_ws;                // scratch: 512*1024 f32 = 2 MB

  gemm_f32_wmma   <<<64,   256, 0, stream>>>(x, T, M);
  pairwise_exp_l1 <<<NB,   256, 0, stream>>>(M, out);
  copy_x          <<<2048, 256, 0, stream>>>(x, out);
}